// MaskedMHA_37855841747088
// MI455X (gfx1250) — compile-verified
//
#include <hip/hip_runtime.h>
#include <hip/hip_bf16.h>

typedef __attribute__((ext_vector_type(2))) float v2f;
typedef __attribute__((ext_vector_type(4))) float v4f;
typedef __attribute__((ext_vector_type(8))) float v8f;

#define N_HEADC 32
#define N_KVC   8
#define HEAD_D  128
#define HID     4096
#define QKV_O   6144          // (8*2+32)*128
#define BSZ     32
#define PASTT   2048
#define TTOT    2049
#define QD      (N_HEADC * HEAD_D)   // 4096
#define KDIM    (N_KVC * HEAD_D)     // 1024

// ---------------------------------------------------------------------------
// Kernel A: fused QKV projection  qkv[32,6144] = x[32,4096] * W^T + b
// One wave per 16-wide N tile; computes both 16-row M tiles (M=32) sharing
// the B operand so W_qkv streams from HBM exactly once.
// Uses V_WMMA_F32_16X16X4_F32 (fp32 matrix core path, no precision loss).
// ---------------------------------------------------------------------------
__global__ __launch_bounds__(256) void qkv_wmma(const float* __restrict__ x,
                                                const float* __restrict__ W,
                                                const float* __restrict__ bias,
                                                float* __restrict__ qkv) {
  const int lane  = threadIdx.x & 31;
  const int wave  = threadIdx.x >> 5;
  const int ntile = blockIdx.x * 8 + wave;        // 0..383
  const int m16   = lane & 15;
  const int koff  = (lane >> 4) << 1;             // 0 or 2 (K pair this lane owns)
  const int col   = ntile * 16 + m16;             // N index this lane owns (B/C/D)

  const float* xr0 = x + (size_t)m16 * HID;        // batches 0..15
  const float* xr1 = x + (size_t)(16 + m16) * HID; // batches 16..31
  const float* wr  = W + (size_t)col * HID;        // B[k][n] = W[n][k]

  v8f c0 = {};
  v8f c1 = {};
  for (int k0 = 0; k0 < HID; k0 += 64) {
    __builtin_prefetch(wr + k0 + 512, 0, 0);       // global_prefetch_b8 ahead on W stream
#pragma unroll
    for (int j = 0; j < 64; j += 4) {
      const int k = k0 + j + koff;
      v2f a0 = *(const v2f*)(xr0 + k);             // A tile rows 0..15
      v2f a1 = *(const v2f*)(xr1 + k);             // A tile rows 16..31
      v2f bb = *(const v2f*)(wr + k);              // shared B operand
      c0 = __builtin_amdgcn_wmma_f32_16x16x4_f32(false, a0, false, bb, (short)0, c0, false, false);
      c1 = __builtin_amdgcn_wmma_f32_16x16x4_f32(false, a1, false, bb, (short)0, c1, false, false);
    }
  }

  const float bv  = bias[col];
  const int   rhi = (lane >> 4) << 3;              // lanes 16-31 hold rows M=8..15
#pragma unroll
  for (int r = 0; r < 8; ++r) {
    const int m = r + rhi;
    qkv[(size_t)m * QKV_O + col]        = c0[r] + bv;
    qkv[(size_t)(16 + m) * QKV_O + col] = c1[r] + bv;
  }
}

// ---------------------------------------------------------------------------
// Kernel B1: bulk KV-cache concat copy (t < 2048), pure streaming memcpy.
// b128 non-temporal loads/stores: ~1 GB of traffic must not thrash L2.
// ---------------------------------------------------------------------------
__global__ void cache_copy(const v4f* __restrict__ ksrc,
                           const v4f* __restrict__ vsrc,
                           v4f* __restrict__ kdst,
                           v4f* __restrict__ vdst) {
  const size_t F4_SRC_B = (size_t)PASTT * KDIM / 4;   // 524288 f4 per batch (src)
  const size_t F4_DST_B = (size_t)TTOT  * KDIM / 4;   // 524544 f4 per batch (dst)
  const size_t NF4      = (size_t)BSZ * F4_SRC_B;     // 16,777,216 per tensor
  const size_t stride   = (size_t)gridDim.x * blockDim.x;
  for (size_t i = (size_t)blockIdx.x * blockDim.x + threadIdx.x; i < 2 * NF4; i += stride) {
    const int    isv = i >= NF4;
    const size_t j   = isv ? (i - NF4) : i;
    const size_t b   = j >> 19;                       // / 524288
    const size_t r   = j & (F4_SRC_B - 1);
    const v4f* s = (isv ? vsrc : ksrc) + b * F4_SRC_B + r;
    v4f*       d = (isv ? vdst : kdst) + b * F4_DST_B + r;
    v4f v = __builtin_nontemporal_load(s);
    __builtin_nontemporal_store(v, d);
  }
}

// ---------------------------------------------------------------------------
// Kernel B2: append the new K/V row (t == 2048) from the QKV result.
// ---------------------------------------------------------------------------
__global__ void cache_append(const float* __restrict__ qkv,
                             float* __restrict__ kdst,
                             float* __restrict__ vdst) {
  const int i   = blockIdx.x * blockDim.x + threadIdx.x;   // 0..65535
  const int isv = i >= BSZ * KDIM;                         // 32768
  const int e   = isv ? (i - BSZ * KDIM) : i;
  const int b   = e >> 10;                                 // / 1024
  const int r   = e & 1023;
  const float val = qkv[(size_t)b * QKV_O + QD + (isv ? KDIM : 0) + r];
  float* dst = isv ? vdst : kdst;
  dst[(size_t)b * TTOT * KDIM + (size_t)PASTT * KDIM + r] = val;
}

// ---------------------------------------------------------------------------
// Kernel C: GQA decode attention. One block per (kv_head, batch) serves the
// 4 q-heads sharing that KV head: K/V read once per block (268 MB each total).
// Phase 1: wave-per-token coalesced float4 dot + wave32 shuffle reduction.
// Phase 2: softmax over T in LDS. Phase 3: coalesced PV with parity split.
// ---------------------------------------------------------------------------
#define STRD 2064
__global__ __launch_bounds__(256) void attn_decode(const float* __restrict__ qkv,
                                                   const float* __restrict__ kc,
                                                   const float* __restrict__ vc,
                                                   const float* __restrict__ mask,
                                                   float* __restrict__ outp) {
  __shared__ float sc[4 * STRD];
  __shared__ float qlds[4 * HEAD_D];
  __shared__ float red[4 * 64];
  __shared__ float stat[8];                 // [0..3]=row max, [4..7]=1/sum
  __shared__ float part[2 * 4 * HEAD_D];

  const int kv   = blockIdx.x;              // 0..7
  const int b    = blockIdx.y;              // 0..31
  const int tid  = threadIdx.x;
  const int lane = tid & 31;
  const int wave = tid >> 5;
  const float scale = 0.08838834764831845f; // 1/sqrt(128)

  // Stage the 4 q-heads (pre-scaled) into LDS.
  for (int i = tid; i < 4 * HEAD_D; i += 256) {
    const int h = kv * 4 + (i >> 7);
    qlds[i] = qkv[(size_t)b * QKV_O + (size_t)h * HEAD_D + (i & 127)] * scale;
  }
  __syncthreads();

  v4f q4[4];
  const v4f* qp = (const v4f*)qlds;
#pragma unroll
  for (int i = 0; i < 4; ++i) q4[i] = qp[i * 32 + lane];

  // Phase 1: scores[h][t] = q_h . k_t  (wave per token; lanes cover D=128).
  const float* kbase = kc + (size_t)b * TTOT * KDIM + (size_t)kv * HEAD_D + lane * 4;
  for (int t = wave; t < TTOT; t += 8) {
    const v4f kvv = *(const v4f*)(kbase + (size_t)t * KDIM);
    float s0 = q4[0].x * kvv.x + q4[0].y * kvv.y + q4[0].z * kvv.z + q4[0].w * kvv.w;
    float s1 = q4[1].x * kvv.x + q4[1].y * kvv.y + q4[1].z * kvv.z + q4[1].w * kvv.w;
    float s2 = q4[2].x * kvv.x + q4[2].y * kvv.y + q4[2].z * kvv.z + q4[2].w * kvv.w;
    float s3 = q4[3].x * kvv.x + q4[3].y * kvv.y + q4[3].z * kvv.z + q4[3].w * kvv.w;
#pragma unroll
    for (int off = 16; off > 0; off >>= 1) {
      s0 += __shfl_xor(s0, off, 32);
      s1 += __shfl_xor(s1, off, 32);
      s2 += __shfl_xor(s2, off, 32);
      s3 += __shfl_xor(s3, off, 32);
    }
    if (lane == 0) {
      const float mk = mask[(size_t)b * TTOT + t];
      sc[0 * STRD + t] = s0 + mk;
      sc[1 * STRD + t] = s1 + mk;
      sc[2 * STRD + t] = s2 + mk;
      sc[3 * STRD + t] = s3 + mk;
    }
  }
  __syncthreads();

  // Phase 2: softmax along T per head (64 threads per head).
  const int h   = tid >> 6;
  const int l64 = tid & 63;
  float lm = -3.4e38f;
  for (int t = l64; t < TTOT; t += 64) lm = fmaxf(lm, sc[h * STRD + t]);
  red[h * 64 + l64] = lm;
  __syncthreads();
  if (l64 == 0) {
    float m = red[h * 64];
    for (int i = 1; i < 64; ++i) m = fmaxf(m, red[h * 64 + i]);
    stat[h] = m;
  }
  __syncthreads();
  const float mh = stat[h];
  float ls = 0.0f;
  for (int t = l64; t < TTOT; t += 64) {
    const float p = __expf(sc[h * STRD + t] - mh);
    sc[h * STRD + t] = p;
    ls += p;
  }
  red[h * 64 + l64] = ls;
  __syncthreads();
  if (l64 == 0) {
    float s = 0.0f;
    for (int i = 0; i < 64; ++i) s += red[h * 64 + i];
    stat[4 + h] = 1.0f / s;
  }
  __syncthreads();

  // Phase 3: out[h][d] = sum_t p[h][t] * v[t][d]; even/odd t split over 2 halves.
  const int d   = tid & 127;
  const int par = tid >> 7;
  float a0 = 0.f, a1 = 0.f, a2 = 0.f, a3 = 0.f;
  const float* vb = vc + (size_t)b * TTOT * KDIM + (size_t)kv * HEAD_D + d;
  for (int t = par; t < TTOT; t += 2) {
    const float v = vb[(size_t)t * KDIM];
    a0 += sc[0 * STRD + t] * v;
    a1 += sc[1 * STRD + t] * v;
    a2 += sc[2 * STRD + t] * v;
    a3 += sc[3 * STRD + t] * v;
  }
  part[par * 512 + 0 * 128 + d] = a0;
  part[par * 512 + 1 * 128 + d] = a1;
  part[par * 512 + 2 * 128 + d] = a2;
  part[par * 512 + 3 * 128 + d] = a3;
  __syncthreads();
  if (tid < 128) {
    float* o = outp + (size_t)b * (N_HEADC * HEAD_D) + (size_t)(kv * 4) * HEAD_D;
#pragma unroll
    for (int i = 0; i < 4; ++i) {
      o[i * 128 + tid] = (part[i * 128 + tid] + part[512 + i * 128 + tid]) * stat[4 + i];
    }
  }
}

extern "C" void kernel_launch(void* const* d_in, const int* in_sizes, int n_in,
                              void* d_out, int out_size, void* d_ws, size_t ws_size,
                              hipStream_t stream) {
  const float* x    = (const float*)d_in[0];   // [32,1,4096]
  const float* kcch = (const float*)d_in[1];   // [32,2048,8,128]
  const float* vcch = (const float*)d_in[2];   // [32,2048,8,128]
  const float* mask = (const float*)d_in[4];   // [32,1,1,2049]
  const float* W    = (const float*)d_in[6];   // [6144,4096]
  const float* bias = (const float*)d_in[7];   // [6144]

  float* out  = (float*)d_out;                                  // [32,32,1,128]
  float* kdst = out + (size_t)BSZ * N_HEADC * HEAD_D;           // k_cat [32,2049,8,128]
  float* vdst = kdst + (size_t)BSZ * TTOT * KDIM;               // v_cat
  float* qkv  = (float*)d_ws;                                   // [32,6144] scratch

  qkv_wmma<<<dim3(48), dim3(256), 0, stream>>>(x, W, bias, qkv);
  cache_copy<<<dim3(8192), dim3(256), 0, stream>>>((const v4f*)kcch, (const v4f*)vcch,
                                                   (v4f*)kdst, (v4f*)vdst);
  cache_append<<<dim3(256), dim3(256), 0, stream>>>(qkv, kdst, vdst);
  attn_decode<<<dim3(8, 32), dim3(256), 0, stream>>>(qkv, kdst, vdst, mask, out);
}